// GCN_Decoder_2104533975392
// MI455X (gfx1250) — compile-verified
//
#include <hip/hip_runtime.h>
#include <hip/hip_bf16.h>
#include <math.h>

typedef float v2f __attribute__((ext_vector_type(2)));
typedef float v8f __attribute__((ext_vector_type(8)));

#define NN   50000
#define NE   1600000
#define D0   16    // E3
#define D1   64    // E4
#define D2   32    // E5
#define LATD 64

// ---------------------------------------------------------------- init
__global__ void k_init(int* cnt, int* cursor) {
    int i = blockIdx.x * blockDim.x + threadIdx.x;
    if (i < NN) { cnt[i] = 0; cursor[i] = 0; }
}

// ------------------------------------------------------- degree count
__global__ void k_degree(const long long* __restrict__ ei, int* __restrict__ cnt) {
    int e = blockIdx.x * blockDim.x + threadIdx.x;
    if (e < NE) {
        int d = (int)ei[NE + e];           // dst row
        atomicAdd(&cnt[d], 1);
    }
}

// -------------------------------------------------- dinv = rsqrt(deg)
__global__ void k_dinv(const int* __restrict__ cnt, float* __restrict__ dinv) {
    int i = blockIdx.x * blockDim.x + threadIdx.x;
    if (i < NN) dinv[i] = rsqrtf((float)(cnt[i] + 1));   // +1 self loop
}

// ----------------------------- exclusive scan of cnt -> rowstart (1 block)
__global__ __launch_bounds__(1024)
void k_scan(const int* __restrict__ cnt, int* __restrict__ rowstart) {
    __shared__ int tmp[1024];
    __shared__ int carry;
    if (threadIdx.x == 0) carry = 0;
    __syncthreads();
    for (int base = 0; base < NN; base += 1024) {
        int i = base + (int)threadIdx.x;
        int v = (i < NN) ? cnt[i] : 0;
        tmp[threadIdx.x] = v;
        __syncthreads();
        for (int off = 1; off < 1024; off <<= 1) {
            int t = (threadIdx.x >= (unsigned)off) ? tmp[threadIdx.x - off] : 0;
            __syncthreads();
            tmp[threadIdx.x] += t;
            __syncthreads();
        }
        if (i < NN) rowstart[i] = carry + tmp[threadIdx.x] - v;  // exclusive
        __syncthreads();
        if (threadIdx.x == 1023) carry += tmp[1023];
        __syncthreads();
    }
}

// ------------------------------------------------------ CSR placement
__global__ void k_place(const long long* __restrict__ ei,
                        const int* __restrict__ rowstart,
                        int* __restrict__ cursor, int* __restrict__ csr) {
    int e = blockIdx.x * blockDim.x + threadIdx.x;
    if (e < NE) {
        int d = (int)ei[NE + e];
        int s = (int)ei[e];
        int pos = atomicAdd(&cursor[d], 1);
        csr[rowstart[d] + pos] = s;
    }
}

// ---------------------------- decoder: h0 = x @ dec_W + dec_b  [N*16]
__global__ __launch_bounds__(256)
void k_decoder(const float* __restrict__ x, const float* __restrict__ decW,
               const float* __restrict__ decb, float* __restrict__ h0) {
    __shared__ float xs[LATD];
    if (threadIdx.x < LATD) xs[threadIdx.x] = x[threadIdx.x];
    __syncthreads();
    int j = blockIdx.x * blockDim.x + threadIdx.x;    // float4 index
    const int J = (NN * D0) / 4;                      // 200000
    if (j >= J) return;
    const float4* W = (const float4*)decW;
    float4 acc = ((const float4*)decb)[j];
    #pragma unroll 8
    for (int l = 0; l < LATD; ++l) {
        float4 wv = W[(size_t)l * J + j];
        float s = xs[l];
        acc.x += s * wv.x; acc.y += s * wv.y;
        acc.z += s * wv.z; acc.w += s * wv.w;
    }
    ((float4*)h0)[j] = acc;
}

// -------- transform 1 (WMMA f32 16x16x4): s1 = dinv[i] * (h0 @ W4)   [N,64]
__global__ __launch_bounds__(32)
void k_xform1(const float* __restrict__ h0, const float* __restrict__ W4,
              const float* __restrict__ dinv, float* __restrict__ s1) {
    int tile = blockIdx.x;              // 16 nodes per wave, 3125 tiles
    int lane = threadIdx.x;
    int l15 = lane & 15, kh = lane >> 4;
    int row0 = tile * 16;
    int row  = row0 + l15;
    // A tiles: 16x4 slices of h0 tile; lane layout per ISA 32-bit A 16x4
    v2f a[4];
    #pragma unroll
    for (int ks = 0; ks < 4; ++ks)
        a[ks] = *(const v2f*)(h0 + row * D0 + 4 * ks + 2 * kh);
    #pragma unroll
    for (int nt = 0; nt < 4; ++nt) {
        v8f c = {};
        #pragma unroll
        for (int ks = 0; ks < 4; ++ks) {
            int k0  = 4 * ks + 2 * kh;
            int col = nt * 16 + l15;
            v2f b;
            b.x = W4[k0 * D1 + col];
            b.y = W4[(k0 + 1) * D1 + col];
            c = __builtin_amdgcn_wmma_f32_16x16x4_f32(
                    false, a[ks], false, b, (short)0, c, false, false);
        }
        #pragma unroll
        for (int r = 0; r < 8; ++r) {
            int m = r + 8 * kh;                       // C/D layout: VGPR r, half
            s1[(size_t)(row0 + m) * D1 + nt * 16 + l15] = dinv[row0 + m] * c[r];
        }
    }
}

// -------- transform 2 (WMMA f32 16x16x4): s2 = dinv[i] * (h1 @ W5)   [N,32]
__global__ __launch_bounds__(32)
void k_xform2(const float* __restrict__ h1, const float* __restrict__ W5,
              const float* __restrict__ dinv, float* __restrict__ s2) {
    int tile = blockIdx.x;
    int lane = threadIdx.x;
    int l15 = lane & 15, kh = lane >> 4;
    int row0 = tile * 16;
    int row  = row0 + l15;
    v2f a[16];
    #pragma unroll
    for (int ks = 0; ks < 16; ++ks)
        a[ks] = *(const v2f*)(h1 + (size_t)row * D1 + 4 * ks + 2 * kh);
    #pragma unroll
    for (int nt = 0; nt < 2; ++nt) {
        v8f c = {};
        #pragma unroll
        for (int ks = 0; ks < 16; ++ks) {
            int k0  = 4 * ks + 2 * kh;
            int col = nt * 16 + l15;
            v2f b;
            b.x = W5[k0 * D2 + col];
            b.y = W5[(k0 + 1) * D2 + col];
            c = __builtin_amdgcn_wmma_f32_16x16x4_f32(
                    false, a[ks], false, b, (short)0, c, false, false);
        }
        #pragma unroll
        for (int r = 0; r < 8; ++r) {
            int m = r + 8 * kh;
            s2[(size_t)(row0 + m) * D2 + nt * 16 + l15] = dinv[row0 + m] * c[r];
        }
    }
}

// ------------- aggregate layer1: h1 = relu(dinv[i]*(sum s1[j] + s1[i]) + b4)
__global__ __launch_bounds__(256)
void k_agg64(const float* __restrict__ s1, const int* __restrict__ rowstart,
             const int* __restrict__ cnt, const int* __restrict__ csr,
             const float* __restrict__ dinv, const float* __restrict__ b4,
             float* __restrict__ h1) {
    int t = blockIdx.x * blockDim.x + threadIdx.x;
    int node = t >> 6, c = t & 63;
    if (node >= NN) return;
    float acc = s1[(size_t)node * D1 + c];           // self loop term
    int s = rowstart[node], e = s + cnt[node];
    for (int p = s; p < e; ++p) {
        int j = csr[p];
        acc += s1[(size_t)j * D1 + c];
    }
    float v = dinv[node] * acc + b4[c];
    h1[(size_t)node * D1 + c] = v > 0.f ? v : 0.f;
}

// ------------- aggregate layer2: h2 = relu(dinv[i]*(sum s2[j] + s2[i]) + b5)
__global__ __launch_bounds__(256)
void k_agg32(const float* __restrict__ s2, const int* __restrict__ rowstart,
             const int* __restrict__ cnt, const int* __restrict__ csr,
             const float* __restrict__ dinv, const float* __restrict__ b5,
             float* __restrict__ h2) {
    int t = blockIdx.x * blockDim.x + threadIdx.x;
    int node = t >> 5, c = t & 31;
    if (node >= NN) return;
    float acc = s2[(size_t)node * D2 + c];
    int s = rowstart[node], e = s + cnt[node];
    for (int p = s; p < e; ++p) {
        int j = csr[p];
        acc += s2[(size_t)j * D2 + c];
    }
    float v = dinv[node] * acc + b5[c];
    h2[(size_t)node * D2 + c] = v > 0.f ? v : 0.f;
}

// ------------- transform 3: s3[i] = dinv[i] * dot(h2[i,:], W6[:,0])
__global__ __launch_bounds__(256)
void k_xform3(const float* __restrict__ h2, const float* __restrict__ W6,
              const float* __restrict__ dinv, float* __restrict__ s3) {
    int i = blockIdx.x * blockDim.x + threadIdx.x;
    if (i >= NN) return;
    const float4* hr = (const float4*)(h2 + (size_t)i * D2);
    const float4* wr = (const float4*)W6;
    float acc = 0.f;
    #pragma unroll
    for (int q = 0; q < D2 / 4; ++q) {
        float4 h = hr[q], w = wr[q];
        acc += h.x * w.x + h.y * w.y + h.z * w.z + h.w * w.w;
    }
    s3[i] = dinv[i] * acc;
}

// ------------- aggregate layer3 + sigmoid -> out[N]
__global__ __launch_bounds__(256)
void k_agg1_sig(const float* __restrict__ s3, const int* __restrict__ rowstart,
                const int* __restrict__ cnt, const int* __restrict__ csr,
                const float* __restrict__ dinv, const float* __restrict__ b6,
                float* __restrict__ out) {
    int i = blockIdx.x * blockDim.x + threadIdx.x;
    if (i >= NN) return;
    float acc = s3[i];
    int s = rowstart[i], e = s + cnt[i];
    for (int p = s; p < e; ++p) acc += s3[csr[p]];
    float v = dinv[i] * acc + b6[0];
    out[i] = 1.f / (1.f + expf(-v));
}

// ================================================================== launch
extern "C" void kernel_launch(void* const* d_in, const int* in_sizes, int n_in,
                              void* d_out, int out_size, void* d_ws, size_t ws_size,
                              hipStream_t stream) {
    const float*     x     = (const float*)d_in[0];
    const long long* ei    = (const long long*)d_in[1];   // int64 [2,E]
    const float*     decW  = (const float*)d_in[2];
    const float*     decb  = (const float*)d_in[3];
    const float*     W4    = (const float*)d_in[4];
    const float*     b4    = (const float*)d_in[5];
    const float*     W5    = (const float*)d_in[6];
    const float*     b5    = (const float*)d_in[7];
    const float*     W6    = (const float*)d_in[8];
    const float*     b6    = (const float*)d_in[9];
    float*           out   = (float*)d_out;

    // workspace carve (256B aligned)
    char* w = (char*)d_ws;
    auto carve = [&](size_t bytes) {
        char* p = w;
        w += (bytes + 255) & ~(size_t)255;
        return p;
    };
    int*   cnt      = (int*)  carve(NN * sizeof(int));
    int*   cursor   = (int*)  carve(NN * sizeof(int));
    int*   rowstart = (int*)  carve(NN * sizeof(int));
    int*   csr      = (int*)  carve((size_t)NE * sizeof(int));
    float* dinv     = (float*)carve(NN * sizeof(float));
    float* h0       = (float*)carve((size_t)NN * D0 * sizeof(float));
    float* s1       = (float*)carve((size_t)NN * D1 * sizeof(float));
    float* h1       = (float*)carve((size_t)NN * D1 * sizeof(float));
    float* s2       = (float*)carve((size_t)NN * D2 * sizeof(float));
    float* h2       = (float*)carve((size_t)NN * D2 * sizeof(float));
    float* s3       = (float*)carve(NN * sizeof(float));
    (void)ws_size; (void)in_sizes; (void)n_in; (void)out_size;

    const int TB = 256;
    const int gN  = (NN + TB - 1) / TB;
    const int gE  = (NE + TB - 1) / TB;

    // graph structure -> CSR
    k_init  <<<gN, TB, 0, stream>>>(cnt, cursor);
    k_degree<<<gE, TB, 0, stream>>>(ei, cnt);
    k_dinv  <<<gN, TB, 0, stream>>>(cnt, dinv);
    k_scan  <<<1, 1024, 0, stream>>>(cnt, rowstart);
    k_place <<<gE, TB, 0, stream>>>(ei, rowstart, cursor, csr);

    // decoder (HBM-bound: streams 204.8 MB of dec_W once)
    k_decoder<<<(NN * D0 / 4 + TB - 1) / TB, TB, 0, stream>>>(x, decW, decb, h0);

    // layer 1
    k_xform1<<<NN / 16, 32, 0, stream>>>(h0, W4, dinv, s1);
    k_agg64 <<<(NN * D1 + TB - 1) / TB, TB, 0, stream>>>(s1, rowstart, cnt, csr, dinv, b4, h1);

    // layer 2
    k_xform2<<<NN / 16, 32, 0, stream>>>(h1, W5, dinv, s2);
    k_agg32 <<<(NN * D2 + TB - 1) / TB, TB, 0, stream>>>(s2, rowstart, cnt, csr, dinv, b5, h2);

    // layer 3 + sigmoid
    k_xform3  <<<gN, TB, 0, stream>>>(h2, W6, dinv, s3);
    k_agg1_sig<<<gN, TB, 0, stream>>>(s3, rowstart, cnt, csr, dinv, b6, out);
}